// GNO_59700045414740
// MI455X (gfx1250) — compile-verified
//
#include <hip/hip_runtime.h>
#include <hip/hip_bf16.h>

// GNO layer fused for MI455X (gfx1250, wave32).
//
// Pipeline:
//   K1 lift_kernel   : h = x @ lift_W + lift_b          (and zero aggr)
//   K2 edge_kernel   : per 16-edge group (one wave32 each):
//                      A[16 edges][256] = edge_attr @ dense_W + dense_b
//                         via 16 N-tiles x 2 chained V_WMMA_F32_16X16X4_F32
//                      stage A in LDS (row stride 260 dwords, conflict-free),
//                      msg[e,i] = sum_j A[e][i*16+j] * h[src[e]][j],
//                      global_atomic_add_f32 into aggr[dst[e]][i]
//   K3 update_kernel : out = tanh(aggr + h@W) @ proj_W + proj_b
//
// WMMA mapping per group: D[m,n] (m = edge-in-group, n = col-in-tile),
//   A_op[m,k] = edge_attr[e0+m, k]   (K padded 6 -> 8, two x4 WMMAs)
//   B_op[k,n] = dense_W[k, t*16+n]   (from LDS)
//   C[m,n]    = dense_b[t*16+n]

#define WPB 2          // waves per block
#define NGRID 2048     // edge-kernel blocks

typedef float v2f __attribute__((ext_vector_type(2)));
typedef float v8f __attribute__((ext_vector_type(8)));

__global__ void __launch_bounds__(256)
lift_kernel(const float* __restrict__ x, const float* __restrict__ liftW,
            const float* __restrict__ liftB, float* __restrict__ h,
            float* __restrict__ aggr, int nNodes) {
    int n = blockIdx.x * blockDim.x + threadIdx.x;
    if (n >= nNodes) return;
    float xi[6];
#pragma unroll
    for (int k = 0; k < 6; ++k) xi[k] = x[(long long)n * 6 + k];
    float hv[16];
#pragma unroll
    for (int i = 0; i < 16; ++i) {
        float acc = liftB[i];
#pragma unroll
        for (int k = 0; k < 6; ++k) acc = fmaf(xi[k], liftW[k * 16 + i], acc);
        hv[i] = acc;
    }
    float4* hp = (float4*)(h + (long long)n * 16);
    float4* ap = (float4*)(aggr + (long long)n * 16);
    float4 z = make_float4(0.f, 0.f, 0.f, 0.f);
#pragma unroll
    for (int q = 0; q < 4; ++q) {
        hp[q] = ((const float4*)hv)[q];
        ap[q] = z;   // zero the scatter accumulator every launch (determinism)
    }
}

__global__ void __launch_bounds__(64)
edge_kernel(const float* __restrict__ edge_attr,
            const long long* __restrict__ edge_index,
            const float* __restrict__ denseW,
            const float* __restrict__ denseB,
            const float* __restrict__ h,
            float* __restrict__ aggr,
            int nEdges) {
    __shared__ float sW[6 * 256];            // dense_W  [k][col]
    __shared__ float sB[256];                // dense_b
    __shared__ float sA[WPB][16 * 260];      // staged A, padded stride 260

    for (int i = threadIdx.x; i < 6 * 256; i += blockDim.x) sW[i] = denseW[i];
    for (int i = threadIdx.x; i < 256; i += blockDim.x) sB[i] = denseB[i];
    __syncthreads();

    const int lane = threadIdx.x & 31;
    const int wave = threadIdx.x >> 5;
    const int n16  = lane & 15;        // N column within tile / edge id for msg
    const int half = lane >> 4;        // 0: lanes 0-15, 1: lanes 16-31
    const int kb   = half * 2;         // WMMA A/B operand K pair base (0 or 2)
    float* Ast = sA[wave];

    const int nGroups = nEdges >> 4;   // 800000 % 16 == 0 for this problem
    const int gStride = gridDim.x * WPB;

    for (int g = blockIdx.x * WPB + wave; g < nGroups; g += gStride) {
        const int e0 = g << 4;

        // ---- A operand: 16 edges x K(6->8) in WMMA 16x4 f32 layout ----
        const float* ea = edge_attr + (long long)(e0 + n16) * 6;
        v2f aLo, aHi;
        aLo.x = ea[kb];                         // K = 0 or 2
        aLo.y = ea[kb + 1];                     // K = 1 or 3
        aHi.x = (kb == 0) ? ea[4] : 0.0f;       // K = 4 (K=6 padded 0)
        aHi.y = (kb == 0) ? ea[5] : 0.0f;       // K = 5 (K=7 padded 0)

#pragma unroll
        for (int t = 0; t < 16; ++t) {
            const int col = t * 16 + n16;       // flat output column
            v2f bLo, bHi;
            bLo.x = sW[(kb)*256 + col];
            bLo.y = sW[(kb + 1) * 256 + col];
            bHi.x = (kb == 0) ? sW[4 * 256 + col] : 0.0f;
            bHi.y = (kb == 0) ? sW[5 * 256 + col] : 0.0f;

            v8f c;
            const float bias = sB[col];
#pragma unroll
            for (int r = 0; r < 8; ++r) c[r] = bias;

            // D = A(16x4) * B(4x16) + C, chained over padded K=8
            c = __builtin_amdgcn_wmma_f32_16x16x4_f32(
                    false, aLo, false, bLo, (short)0, c, false, false);
            c = __builtin_amdgcn_wmma_f32_16x16x4_f32(
                    false, aHi, false, bHi, (short)0, c, false, false);

            // stage tile: rows m = r + 8*half, column = col
#pragma unroll
            for (int r = 0; r < 8; ++r)
                Ast[(r + half * 8) * 260 + col] = c[r];
        }

        // wave-local DS fence before re-reading the staged A block
        __builtin_amdgcn_wave_barrier();
        asm volatile("s_wait_dscnt 0" ::: "memory");
        __builtin_amdgcn_wave_barrier();

        // ---- msg + scatter: 2 lanes per edge, 8 rows each ----
        const long long e = e0 + n16;
        const long long s = edge_index[e];
        const long long d = edge_index[(long long)nEdges + e];

        float hs[16];
        const float4* hp = (const float4*)(h + s * 16);
#pragma unroll
        for (int q = 0; q < 4; ++q) ((float4*)hs)[q] = hp[q];

#pragma unroll
        for (int r = 0; r < 8; ++r) {
            const int i = half * 8 + r;
            float av[16];
            const float4* ar = (const float4*)(Ast + n16 * 260 + i * 16);
#pragma unroll
            for (int q = 0; q < 4; ++q) ((float4*)av)[q] = ar[q];
            float acc = 0.0f;
#pragma unroll
            for (int j = 0; j < 16; ++j) acc = fmaf(av[j], hs[j], acc);
            unsafeAtomicAdd(&aggr[d * 16 + i], acc);  // global_atomic_add_f32
        }
        __builtin_amdgcn_wave_barrier();
    }
}

__global__ void __launch_bounds__(256)
update_kernel(const float* __restrict__ h, const float* __restrict__ aggr,
              const float* __restrict__ W, const float* __restrict__ projW,
              const float* __restrict__ projB, float* __restrict__ out,
              int nNodes) {
    int n = blockIdx.x * blockDim.x + threadIdx.x;
    if (n >= nNodes) return;
    float hv[16], av[16];
    const float4* hp = (const float4*)(h + (long long)n * 16);
    const float4* ap = (const float4*)(aggr + (long long)n * 16);
#pragma unroll
    for (int q = 0; q < 4; ++q) {
        ((float4*)hv)[q] = hp[q];
        ((float4*)av)[q] = ap[q];
    }
    float o = 0.0f;
#pragma unroll
    for (int i = 0; i < 16; ++i) {
        float u = av[i];
#pragma unroll
        for (int j = 0; j < 16; ++j) u = fmaf(hv[j], W[j * 16 + i], u);
        o = fmaf(tanhf(u), projW[i], o);
    }
    out[n] = o + projB[0];
}

extern "C" void kernel_launch(void* const* d_in, const int* in_sizes, int n_in,
                              void* d_out, int out_size, void* d_ws, size_t ws_size,
                              hipStream_t stream) {
    const float*     x          = (const float*)d_in[0];
    const long long* edge_index = (const long long*)d_in[1];  // int64 [2,E]
    const float*     edge_attr  = (const float*)d_in[2];
    const float*     liftW      = (const float*)d_in[3];
    const float*     liftB      = (const float*)d_in[4];
    const float*     denseW     = (const float*)d_in[5];
    const float*     denseB     = (const float*)d_in[6];
    const float*     W          = (const float*)d_in[7];
    const float*     projW      = (const float*)d_in[8];
    const float*     projB      = (const float*)d_in[9];
    float* out = (float*)d_out;

    const int nNodes = in_sizes[0] / 6;   // 50000
    const int nEdges = in_sizes[2] / 6;   // 800000

    float* h    = (float*)d_ws;                       // [nNodes][16]
    float* aggr = h + (size_t)nNodes * 16;            // [nNodes][16]

    dim3 nb(256), ng((nNodes + 255) / 256);
    lift_kernel<<<ng, nb, 0, stream>>>(x, liftW, liftB, h, aggr, nNodes);
    edge_kernel<<<dim3(NGRID), dim3(64), 0, stream>>>(
        edge_attr, edge_index, denseW, denseB, h, aggr, nEdges);
    update_kernel<<<ng, nb, 0, stream>>>(h, aggr, W, projW, projB, out, nNodes);
}